// Model_58609123721280
// MI455X (gfx1250) — compile-verified
//
#include <hip/hip_runtime.h>

// Native vector type so __builtin_nontemporal_load/store accept it
// (HIP's float4 class type is not guaranteed to work with the builtin).
typedef float v4f __attribute__((ext_vector_type(4)));

// x: [B=8, D1=1024, D2=1024, C=4] f32. out: [B, D1, D2] f32.
// out[b,i,j] = (x[b,i,j,3] > 0.5f) ? (float)i : 0.0f
//
// Each thread produces 4 consecutive outputs (one b128 NT store) and
// consumes 4 float4s (four b128 NT loads). Since D2 == 1024 is a
// multiple of 4, an aligned group of 4 outputs shares one row index.

#ifndef D1_DIM
#define D1_DIM 1024
#endif
#ifndef D2_DIM
#define D2_DIM 1024
#endif

__global__ __launch_bounds__(256) void mask_rowidx_kernel(
    const v4f* __restrict__ x,   // input viewed as float4 per output element
    float* __restrict__ out,
    int n_out)                   // total output elements (B*D1*D2)
{
    const int tid  = blockIdx.x * blockDim.x + threadIdx.x;
    const int base = tid * 4;    // first output element this thread owns
    if (base >= n_out) return;

    if (base + 4 <= n_out) {
        // Fast path: 4 aligned outputs, same D1 row (D2 % 4 == 0).
        const float row = (float)((base / D2_DIM) % D1_DIM);

        v4f a = __builtin_nontemporal_load(&x[base + 0]);
        v4f b = __builtin_nontemporal_load(&x[base + 1]);
        v4f c = __builtin_nontemporal_load(&x[base + 2]);
        v4f d = __builtin_nontemporal_load(&x[base + 3]);

        v4f r;
        r.x = (a.w > 0.5f) ? row : 0.0f;
        r.y = (b.w > 0.5f) ? row : 0.0f;
        r.z = (c.w > 0.5f) ? row : 0.0f;
        r.w = (d.w > 0.5f) ? row : 0.0f;

        __builtin_nontemporal_store(r, (v4f*)(out + base));
    } else {
        // Tail (not hit for the reference shape, kept for safety).
        for (int i = base; i < n_out; ++i) {
            const float row = (float)((i / D2_DIM) % D1_DIM);
            v4f a = __builtin_nontemporal_load(&x[i]);
            out[i] = (a.w > 0.5f) ? row : 0.0f;
        }
    }
}

extern "C" void kernel_launch(void* const* d_in, const int* in_sizes, int n_in,
                              void* d_out, int out_size, void* d_ws, size_t ws_size,
                              hipStream_t stream) {
    (void)n_in; (void)d_ws; (void)ws_size;

    const v4f* x   = (const v4f*)d_in[0];
    float*     out = (float*)d_out;

    // in_sizes[0] = B*D1*D2*C flat floats; C == 4 -> output element count.
    const int n_out = in_sizes[0] / 4;   // == out_size

    const int per_thread = 4;
    const int threads    = (n_out + per_thread - 1) / per_thread;
    const int block      = 256;                       // 8 wave32s
    const int grid       = (threads + block - 1) / block;

    mask_rowidx_kernel<<<grid, block, 0, stream>>>(x, out, n_out);
}